// EGNN_17626545783200
// MI455X (gfx1250) — compile-verified
//
#include <hip/hip_runtime.h>

#define N_FEATS 512
#define HID     128

typedef __attribute__((ext_vector_type(2))) float v2f;
typedef __attribute__((ext_vector_type(8))) float v8f;

// ---------------------------------------------------------------------------
// Graph-norm helpers
// ---------------------------------------------------------------------------
__global__ void degree_kernel(const int* __restrict__ dst, float* __restrict__ deg, int E) {
    int e = blockIdx.x * blockDim.x + threadIdx.x;
    if (e < E) atomicAdd(&deg[dst[e]], 1.0f);
}

// deg -> dinv in place; +1 accounts for the self-loop added by gcn_norm
__global__ void finalize_dinv_kernel(float* __restrict__ deg, int n) {
    int i = blockIdx.x * blockDim.x + threadIdx.x;
    if (i < n) deg[i] = rsqrtf(deg[i] + 1.0f);
}

// pre = 0.8*dinv^2*h  (self-loop term of agg)  + 0.1*h + 0.1*x0
// one wave per node, float4 per lane (32*4 = 128 features), fully coalesced
__global__ void init_pre_kernel(const float* __restrict__ h, const float* __restrict__ x0,
                                const float* __restrict__ dinv, float* __restrict__ pre, int n) {
    int t    = blockIdx.x * blockDim.x + threadIdx.x;
    int node = t >> 5;
    int c4   = (t & 31) << 2;
    if (node >= n) return;
    float di = dinv[node];
    float sw = 0.8f * di * di + 0.1f;          // self-loop + residual on h
    const float4 hv = *reinterpret_cast<const float4*>(h  + (size_t)node * HID + c4);
    const float4 xv = *reinterpret_cast<const float4*>(x0 + (size_t)node * HID + c4);
    float4 o;
    o.x = sw * hv.x + 0.1f * xv.x;
    o.y = sw * hv.y + 0.1f * xv.y;
    o.z = sw * hv.z + 0.1f * xv.z;
    o.w = sw * hv.w + 0.1f * xv.w;
    *reinterpret_cast<float4*>(pre + (size_t)node * HID + c4) = o;
}

// one wave32 per edge: gather h[src] (512B coalesced), scatter 0.8*norm*h into pre[dst]
__global__ void edge_agg_kernel(const int* __restrict__ src, const int* __restrict__ dst,
                                const float* __restrict__ dinv, const float* __restrict__ h,
                                float* __restrict__ pre, int E) {
    long long gt = (long long)blockIdx.x * blockDim.x + threadIdx.x;
    int e    = (int)(gt >> 5);
    int lane = (int)(gt & 31);
    if (e >= E) return;
    int s = src[e];
    int d = dst[e];
    float w = 0.8f * dinv[s] * dinv[d];
    const float4 hv = *reinterpret_cast<const float4*>(h + (size_t)s * HID + (lane << 2));
    float* p = pre + (size_t)d * HID + (lane << 2);
    atomicAdd(p + 0, w * hv.x);
    atomicAdd(p + 1, w * hv.y);
    atomicAdd(p + 2, w * hv.z);
    atomicAdd(p + 3, w * hv.w);
}

// ---------------------------------------------------------------------------
// WMMA fp32 GEMM:  C[M,Nact] = act( A[M,K] @ B[K,Nact] + bias )
//   128-row block tile, 8 waves; wave w owns rows [w*16, w*16+16), NT n-tiles.
//   K consumed in chunks of 16 via LDS, 4x v_wmma_f32_16x16x4_f32 per chunk.
//   Both A and B^T panels use a 20-float row stride:
//     * rows 16B-aligned, (K,K+1) pairs 8B-aligned -> single ds_load_b64/frag
//     * 20*r mod 64 banks distinct for r=0..15; +2-offset half-wave cannot
//       collide (10*delta == 1 mod 32 has no solution) -> conflict-free wave32
//   ACT: 0 = relu(y+bias), 1 = max(y,bias) [SReLU], 2 = y+bias
// ---------------------------------------------------------------------------
template <int K, int NT, int ACT, bool DUAL>
__global__ __launch_bounds__(256) void wmma_gemm_kernel(
        const float* __restrict__ A, const float* __restrict__ B,
        const float* __restrict__ bias, float* __restrict__ C,
        float* __restrict__ C2, int M, int Nact) {
    constexpr int NPAD = NT * 16;
    constexpr int LS   = 20;                 // LDS row stride in floats
    __shared__ float As [128][LS];           // As[row][k]   (k = 0..15)
    __shared__ float BsT[NPAD][LS];          // BsT[col][k]  (k = 0..15)

    const int tid  = threadIdx.x;
    const int wave = tid >> 5;
    const int lane = tid & 31;
    const int lr   = lane & 15;
    const int half = lane >> 4;              // 0: lanes 0-15, 1: lanes 16-31
    const int m0   = blockIdx.x * 128;
    const int wrow = wave * 16;

    v8f acc[NT] = {};

    for (int k0 = 0; k0 < K; k0 += 16) {
        // ---- stage A panel: 128 x 16, float4 global loads, 16B LDS stores ----
#pragma unroll
        for (int i = 0; i < 2; ++i) {
            int fidx = tid + i * 256;        // 0..511 float4s
            int r    = fidx >> 2;
            int c4   = (fidx & 3) << 2;
            float4 v = make_float4(0.f, 0.f, 0.f, 0.f);
            int row  = m0 + r;
            if (row < M)
                v = *reinterpret_cast<const float4*>(A + (size_t)row * K + k0 + c4);
            As[r][c4 + 0] = v.x; As[r][c4 + 1] = v.y;
            As[r][c4 + 2] = v.z; As[r][c4 + 3] = v.w;
        }
        // ---- stage B panel transposed: BsT[col][k], coalesced global reads ----
        for (int idx = tid; idx < 16 * NPAD; idx += 256) {
            int kk = idx / NPAD;
            int c  = idx % NPAD;
            float v = 0.f;
            if (c < Nact) v = B[(size_t)(k0 + kk) * Nact + c];
            BsT[c][kk] = v;
        }
        __syncthreads();

        // ---- 4 WMMA k-steps of 4: pure ds_load_b64 -> v_wmma ----
#pragma unroll
        for (int ks = 0; ks < 4; ++ks) {
            const int kb = ks * 4 + 2 * half;          // even -> 8B aligned
            v2f a = *reinterpret_cast<const v2f*>(&As[wrow + lr][kb]);
#pragma unroll
            for (int nt = 0; nt < NT; ++nt) {
                v2f b = *reinterpret_cast<const v2f*>(&BsT[nt * 16 + lr][kb]);
                acc[nt] = __builtin_amdgcn_wmma_f32_16x16x4_f32(
                    false, a, false, b, (short)0, acc[nt], false, false);
            }
        }
        __syncthreads();
    }

    // ---- epilogue: C/D layout -> vgpr r = row (half*8 + r), col = lane&15 ----
#pragma unroll
    for (int nt = 0; nt < NT; ++nt) {
        const int n = nt * 16 + lr;
        if (n >= Nact) continue;
        const float bv = bias[n];
#pragma unroll
        for (int r = 0; r < 8; ++r) {
            const int m = m0 + wrow + half * 8 + r;
            if (m < M) {
                float v = acc[nt][r];
                if (ACT == 0)      v = fmaxf(v + bv, 0.0f);
                else if (ACT == 1) v = fmaxf(v, bv);
                else               v = v + bv;
                C[(size_t)m * Nact + n] = v;
                if (DUAL) C2[(size_t)m * Nact + n] = v;
            }
        }
    }
}

// ---------------------------------------------------------------------------
extern "C" void kernel_launch(void* const* d_in, const int* in_sizes, int n_in,
                              void* d_out, int out_size, void* d_ws, size_t ws_size,
                              hipStream_t stream) {
    const float* x     = (const float*)d_in[0];
    const int*   edges = (const int*)d_in[1];
    const float* W_in  = (const float*)d_in[2];
    const float* b_in  = (const float*)d_in[3];
    const float* W_lay = (const float*)d_in[4];
    const float* sbias = (const float*)d_in[5];
    const float* W_out = (const float*)d_in[6];
    const float* b_out = (const float*)d_in[7];
    float* out = (float*)d_out;

    const int M = in_sizes[0] / N_FEATS;          // 50000 nodes
    const int E = in_sizes[1] / 2;                // 800000 edges
    const int L = in_sizes[4] / (HID * HID);      // 4 layers
    const int Ncls = in_sizes[7];                 // 40 classes

    // workspace layout (16B aligned slices)
    size_t mpad = ((size_t)M + 3) & ~(size_t)3;
    float* dinv = (float*)d_ws;                   // M
    float* h    = dinv + mpad;                    // M*HID
    float* x0   = h    + (size_t)M * HID;         // M*HID
    float* pre  = x0   + (size_t)M * HID;         // M*HID

    const int* src = edges;
    const int* dst = edges + E;

    // gcn_norm: degree (+self-loop) -> dinv
    hipMemsetAsync(dinv, 0, sizeof(float) * M, stream);
    degree_kernel<<<(E + 255) / 256, 256, 0, stream>>>(dst, dinv, E);
    finalize_dinv_kernel<<<(M + 255) / 256, 256, 0, stream>>>(dinv, M);

    // input projection: h = x0 = relu(x @ W_in + b_in)
    dim3 gblk((M + 127) / 128);
    wmma_gemm_kernel<N_FEATS, HID / 16, 0, true>
        <<<gblk, 256, 0, stream>>>(x, W_in, b_in, h, x0, M, HID);

    const int initBlocks = (M * 32 + 255) / 256;
    const int edgeBlocks = (int)(((long long)E * 32 + 255) / 256);
    for (int l = 0; l < L; ++l) {
        init_pre_kernel<<<initBlocks, 256, 0, stream>>>(h, x0, dinv, pre, M);
        edge_agg_kernel<<<edgeBlocks, 256, 0, stream>>>(src, dst, dinv, h, pre, E);
        // h = SReLU(pre @ W_l)  where SReLU(y) = max(y, srelu_bias)
        wmma_gemm_kernel<HID, HID / 16, 1, false>
            <<<gblk, 256, 0, stream>>>(pre, W_lay + (size_t)l * HID * HID,
                                       sbias + (size_t)l * HID, h, nullptr, M, HID);
    }

    // output: out = h @ W_out + b_out  (N=40, padded to 48 -> 3 n-tiles, guarded)
    wmma_gemm_kernel<HID, 3, 2, false>
        <<<gblk, 256, 0, stream>>>(h, W_out, b_out, out, nullptr, M, Ncls);
}